// SimplicialFeatLearningLayerTri_21217138442473
// MI455X (gfx1250) — compile-verified
//
#include <hip/hip_runtime.h>
#include <hip/hip_bf16.h>

typedef __attribute__((ext_vector_type(2))) float v2f;
typedef __attribute__((ext_vector_type(8))) float v8f;

#define NPTS 1024
#define DF 32
#define NBW 16
#define SA 36              // A-tile LDS row stride (floats): 16B-aligned, conflict-free b64 frag reads
#define SB 136             // B-tile LDS row stride (floats): 16B-aligned, halves on disjoint banks
#define TILE 128

__device__ __forceinline__ v8f wmma_f32(v2f a, v2f b, v8f c) {
  // D = A(16x4 f32) x B(4x16 f32) + C(16x16 f32)
  return __builtin_amdgcn_wmma_f32_16x16x4_f32(false, a, false, b, (short)0, c, false, false);
}

// LDS byte offset of a generic pointer into __shared__ (flat LDS aperture keeps offset in addr[31:0])
__device__ __forceinline__ uint32_t lds_off(const void* p) {
  return (uint32_t)(uintptr_t)p;
}

// async copy 16 bytes global -> LDS, tracked by ASYNCcnt
__device__ __forceinline__ void async_b128(uint32_t dst_lds, const float* src) {
  asm volatile("global_load_async_to_lds_b128 %0, %1, off" :: "v"(dst_lds), "v"(src) : "memory");
}
__device__ __forceinline__ void wait_async_prefetch() {   // current step done, next step may be in flight
  asm volatile("s_wait_asynccnt 8" ::: "memory");
}
__device__ __forceinline__ void wait_async_all() {
  asm volatile("s_wait_asynccnt 0" ::: "memory");
}

// ---------------- X = pc * alpha, sq = ||x||^2 ----------------
__global__ __launch_bounds__(256) void k_xsq(const float* __restrict__ pc,
                                             const float* __restrict__ alphas,
                                             float* __restrict__ Xall,
                                             float* __restrict__ sqall) {
  int g = blockIdx.x * 256 + threadIdx.x;          // 0 .. 16383  (bw * 1024 + n)
  int bw = g >> 10, n = g & 1023;
  int b = bw >> 2, w = bw & 3;
  const float* src = pc + ((size_t)b * NPTS + n) * DF;
  const float* al  = alphas + w * DF;
  float* dst = Xall + (size_t)g * DF;
  float s = 0.f;
#pragma unroll
  for (int d = 0; d < DF; ++d) {
    float x = src[d] * al[d];
    dst[d] = x;
    s += x * x;
  }
  sqall[g] = s;
}

// ---------------- Gram via fp32 WMMA -> thresholded kernel matrix W ----------------
__global__ __launch_bounds__(256) void k_gram(const float* __restrict__ Xall,
                                              const float* __restrict__ sqall,
                                              const float* __restrict__ sigmap,
                                              float* __restrict__ Wall) {
  int bw = blockIdx.z;
  const float* X  = Xall  + (size_t)bw * NPTS * DF;
  const float* sq = sqall + (size_t)bw * NPTS;
  float* W = Wall + (size_t)bw * NPTS * NPTS;
  int m0 = blockIdx.y * TILE, n0 = blockIdx.x * TILE;

  __shared__ float sAm[TILE * SA];
  __shared__ float sBn[TILE * SA];
  int t = threadIdx.x;
#pragma unroll
  for (int i = 0; i < 4; ++i) {                    // 1024 float4 = 4096 floats per tile
    int idx = t + i * 256;
    int r = idx >> 3, kq = (idx & 7) << 2;
    *(float4*)&sAm[r * SA + kq] = *(const float4*)&X[(size_t)(m0 + r) * DF + kq];
    *(float4*)&sBn[r * SA + kq] = *(const float4*)&X[(size_t)(n0 + r) * DF + kq];
  }
  __syncthreads();

  int lane = t & 31, wid = t >> 5;
  int wm = wid & 3, wn = wid >> 2;                 // 4x2 wave grid, each wave 32x64
  int lhalf = lane >> 4, l15 = lane & 15;
  v8f acc[2][4];
#pragma unroll
  for (int mi = 0; mi < 2; ++mi)
#pragma unroll
    for (int ni = 0; ni < 4; ++ni) acc[mi][ni] = (v8f){0.f,0.f,0.f,0.f,0.f,0.f,0.f,0.f};

#pragma unroll
  for (int kk = 0; kk < DF; kk += 4) {
    int lk = kk + lhalf * 2;
    v2f af[2], bf[4];
#pragma unroll
    for (int mi = 0; mi < 2; ++mi)
      af[mi] = *(const v2f*)&sAm[(wm * 32 + mi * 16 + l15) * SA + lk];
#pragma unroll
    for (int ni = 0; ni < 4; ++ni)
      bf[ni] = *(const v2f*)&sBn[(wn * 64 + ni * 16 + l15) * SA + lk];
#pragma unroll
    for (int mi = 0; mi < 2; ++mi)
#pragma unroll
      for (int ni = 0; ni < 4; ++ni)
        acc[mi][ni] = wmma_f32(af[mi], bf[ni], acc[mi][ni]);
  }

  float sigma = *sigmap;
#pragma unroll
  for (int mi = 0; mi < 2; ++mi)
#pragma unroll
    for (int ni = 0; ni < 4; ++ni) {
      int col = n0 + wn * 64 + ni * 16 + l15;
      float sqc = sq[col];
#pragma unroll
      for (int r = 0; r < 8; ++r) {
        int row = m0 + wm * 32 + mi * 16 + lhalf * 8 + r;
        float g = acc[mi][ni][r];
        float dm = sq[row] + sqc - 2.0f * g;
        float kv = __expf(-dm / sigma);
        W[(size_t)row * NPTS + col] = (kv >= 0.5f) ? kv : 0.f;
      }
    }
}

// ---------------- P = 0.5 (I + W / deg), one block per row ----------------
__global__ __launch_bounds__(256) void k_prow(const float* __restrict__ Wall,
                                              float* __restrict__ Pall) {
  int bw = blockIdx.y, row = blockIdx.x;
  const float* Wr = Wall + ((size_t)bw * NPTS + row) * NPTS;
  float* Pr = Pall + ((size_t)bw * NPTS + row) * NPTS;
  __shared__ float red[256];
  int t = threadIdx.x;
  float s = 0.f;
  for (int i = t; i < NPTS; i += 256) s += Wr[i];
  red[t] = s;
  __syncthreads();
  for (int off = 128; off >= 1; off >>= 1) {
    if (t < off) red[t] += red[t + off];
    __syncthreads();
  }
  float invdeg = 1.0f / red[0];
  for (int i = t; i < NPTS; i += 256)
    Pr[i] = 0.5f * (Wr[i] * invdeg + (i == row ? 1.0f : 0.0f));
}

// ---------------- C = A * A  (1024^3, fp32 WMMA, async double-buffered LDS) ----------------
__global__ __launch_bounds__(256) void k_square(const float* __restrict__ Aall,
                                                float* __restrict__ Call) {
  int bw = blockIdx.z;
  const float* A = Aall + (size_t)bw * NPTS * NPTS;
  float* C = Call + (size_t)bw * NPTS * NPTS;
  int m0 = blockIdx.y * TILE, n0 = blockIdx.x * TILE;

  __shared__ float sA[2][TILE * SA];               // A rows [m][k], row-major, stride SA
  __shared__ float sB[2][32 * SB];                 // B rows [k][c], row-major, stride SB
  int t = threadIdx.x, lane = t & 31, wid = t >> 5;
  int wm = wid & 3, wn = wid >> 2;                 // 4x2 wave grid, each wave 32x64
  int lhalf = lane >> 4, l15 = lane & 15;

  // per-thread async-copy coordinates (16B granules)
  int ar = (t * 4) >> 5;                           // via idx=t+i*256: r = idx>>3
  int bk = (t * 4) >> 7;                           // via idx=t+i*256: k = idx>>5

  v8f acc[2][4];
#pragma unroll
  for (int mi = 0; mi < 2; ++mi)
#pragma unroll
    for (int ni = 0; ni < 4; ++ni) acc[mi][ni] = (v8f){0.f,0.f,0.f,0.f,0.f,0.f,0.f,0.f};

  // issue async copies for K-step k0 into buffer p (8 instructions per wave)
  auto issue = [&](int p, int k0) {
#pragma unroll
    for (int i = 0; i < 4; ++i) {                  // A tile: 128 x 32 floats = 1024 float4
      int idx = t + i * 256;
      int r = idx >> 3, kq = (idx & 7) << 2;
      async_b128(lds_off(&sA[p][r * SA + kq]),
                 &A[(size_t)(m0 + r) * NPTS + k0 + kq]);
    }
#pragma unroll
    for (int i = 0; i < 4; ++i) {                  // B tile: 32 x 128 floats = 1024 float4
      int idx = t + i * 256;
      int k = idx >> 5, c4 = (idx & 31) << 2;
      async_b128(lds_off(&sB[p][k * SB + c4]),
                 &A[(size_t)(k0 + k) * NPTS + n0 + c4]);
    }
  };
  (void)ar; (void)bk;

  issue(0, 0);
  for (int s = 0; s < NPTS / 32; ++s) {
    int p = s & 1;
    bool more = (s + 1) < (NPTS / 32);
    if (more) issue(p ^ 1, (s + 1) * 32);
    if (more) wait_async_prefetch(); else wait_async_all();
    __syncthreads();

    const float* sAp = &sA[p][0];
    const float* sBp = &sB[p][0];
#pragma unroll
    for (int kk = 0; kk < 32; kk += 4) {
      int lk = kk + lhalf * 2;
      v2f af[2], bf[4];
#pragma unroll
      for (int mi = 0; mi < 2; ++mi)
        af[mi] = *(const v2f*)&sAp[(wm * 32 + mi * 16 + l15) * SA + lk];
#pragma unroll
      for (int ni = 0; ni < 4; ++ni) {
        const float* bp = &sBp[lk * SB + (wn * 64 + ni * 16 + l15)];
        v2f b; b.x = bp[0]; b.y = bp[SB];
        bf[ni] = b;
      }
#pragma unroll
      for (int mi = 0; mi < 2; ++mi)
#pragma unroll
        for (int ni = 0; ni < 4; ++ni)
          acc[mi][ni] = wmma_f32(af[mi], bf[ni], acc[mi][ni]);
    }
    __syncthreads();
  }

#pragma unroll
  for (int mi = 0; mi < 2; ++mi)
#pragma unroll
    for (int ni = 0; ni < 4; ++ni) {
      int col = n0 + wn * 64 + ni * 16 + l15;
#pragma unroll
      for (int r = 0; r < 8; ++r) {
        int row = m0 + wm * 32 + mi * 16 + lhalf * 8 + r;
        C[(size_t)row * NPTS + col] = acc[mi][ni][r];
      }
    }
}

// ---------------- Fout = | (Pa - Pb) * Fin |, one block per output row ----------------
__global__ __launch_bounds__(256) void k_apply(const float* __restrict__ Pa,
                                               const float* __restrict__ Pb,
                                               const float* __restrict__ Fin,
                                               float* __restrict__ Fout) {
  int bw = blockIdx.y, n = blockIdx.x;
  const float* pa = Pa + ((size_t)bw * NPTS + n) * NPTS;
  const float* pb = Pb + ((size_t)bw * NPTS + n) * NPTS;
  const float* F  = Fin + (size_t)bw * NPTS * DF;
  int t = threadIdx.x, d = t & 31, ks = t >> 5;    // 8 k-slices of 128
  float s = 0.f;
  int kbeg = ks * 128;
  for (int k = kbeg; k < kbeg + 128; ++k) {
    float m = pa[k] - pb[k];
    s += m * F[(size_t)k * DF + d];
  }
  __shared__ float red[256];
  red[t] = s;
  __syncthreads();
  for (int off = 4; off >= 1; off >>= 1) {
    if (ks < off) red[t] += red[t + off * 32];
    __syncthreads();
  }
  if (t < 32)
    Fout[((size_t)bw * NPTS + n) * DF + d] = fabsf(red[d]);
}

// ---------------- mean pool over N, write to output ----------------
__global__ __launch_bounds__(256) void k_pool(const float* __restrict__ feats,
                                              float* __restrict__ out) {
  int bw = blockIdx.x, arr = blockIdx.y;           // arr: 0=X 1=f0 2=f1 3=f2 4=s01 5=s02 6=s12
  const float* F = feats + ((size_t)arr * NBW + bw) * NPTS * DF;
  int t = threadIdx.x, d = t & 31, ks = t >> 5;
  float s = 0.f;
  int nbeg = ks * 128;
  for (int n = nbeg; n < nbeg + 128; ++n) s += F[(size_t)n * DF + d];
  __shared__ float red[256];
  red[t] = s;
  __syncthreads();
  for (int off = 4; off >= 1; off >>= 1) {
    if (ks < off) red[t] += red[t + off * 32];
    __syncthreads();
  }
  if (t < 32) {
    int b = bw >> 2, w = bw & 3;
    out[(size_t)b * 896 + w * 224 + arr * 32 + d] = red[d] * (1.0f / 1024.0f);
  }
}

extern "C" void kernel_launch(void* const* d_in, const int* in_sizes, int n_in,
                              void* d_out, int out_size, void* d_ws, size_t ws_size,
                              hipStream_t stream) {
  (void)in_sizes; (void)n_in; (void)out_size; (void)ws_size;
  const float* pc     = (const float*)d_in[0];     // [4,1024,32]
  const float* alphas = (const float*)d_in[1];     // [4,32]
  const float* sigma  = (const float*)d_in[2];     // [1]
  float* out = (float*)d_out;                      // [4,896]
  float* ws  = (float*)d_ws;

  // workspace layout (floats)
  const size_t FEAT = (size_t)NBW * NPTS * DF;     // 524288 per feature array
  float* Xf  = ws;                                 // arr 0 (X) .. arr 6 (s12) contiguous
  float* f0  = ws + 1 * FEAT;
  float* f1  = ws + 2 * FEAT;
  float* f2  = ws + 3 * FEAT;
  float* s01 = ws + 4 * FEAT;
  float* s02 = ws + 5 * FEAT;
  float* s12 = ws + 6 * FEAT;
  float* sq  = ws + 7 * FEAT;                      // 16*1024
  float* M0  = sq + (size_t)NBW * NPTS;            // 16 * 1024*1024
  float* M1  = M0 + (size_t)NBW * NPTS * NPTS;

  dim3 blk(256);
  dim3 gGemm(NPTS / TILE, NPTS / TILE, NBW);       // (8,8,16)
  dim3 gRow(NPTS, NBW);                            // (1024,16)
  dim3 gPool(NBW, 7);

  // 1) X, squared norms
  k_xsq<<<dim3(NBW * NPTS / 256), blk, 0, stream>>>(pc, alphas, Xf, sq);
  // 2) W (thresholded Gaussian kernel) into M1
  k_gram<<<gGemm, blk, 0, stream>>>(Xf, sq, sigma, M1);
  // 3) P into M0
  k_prow<<<gRow, blk, 0, stream>>>(M1, M0);
  // 4) P2 = P*P  (M1)
  k_square<<<gGemm, blk, 0, stream>>>(M0, M1);
  // 5) f0 = |(P - P2) X|
  k_apply<<<gRow, blk, 0, stream>>>(M0, M1, Xf, f0);
  // 6) P4 = P2*P2 (M0)
  k_square<<<gGemm, blk, 0, stream>>>(M1, M0);
  // 7) f1 = |(P2 - P4) X| ; s01 = |(P2 - P4) f0|
  k_apply<<<gRow, blk, 0, stream>>>(M1, M0, Xf, f1);
  k_apply<<<gRow, blk, 0, stream>>>(M1, M0, f0, s01);
  // 8) P8 = P4*P4 (M1)
  k_square<<<gGemm, blk, 0, stream>>>(M0, M1);
  // 9) f2 = |(P4 - P8) X| ; s02 = |(P4 - P8) f0| ; s12 = |(P4 - P8) f1|
  k_apply<<<gRow, blk, 0, stream>>>(M0, M1, Xf, f2);
  k_apply<<<gRow, blk, 0, stream>>>(M0, M1, f0, s02);
  k_apply<<<gRow, blk, 0, stream>>>(M0, M1, f1, s12);
  // 10) mean pool -> out
  k_pool<<<gPool, blk, 0, stream>>>(Xf, out);
}